// RegressionARMA_12189117186554
// MI455X (gfx1250) — compile-verified
//
#include <hip/hip_runtime.h>

// Problem constants (match reference)
#define N_NODES 100000
#define N_EDGES 1600000
#define F_INF   128
#define H_F     16
#define OUT_F   10
#define K_ST    3
#define PADW    16   // padded feature width for all [K,N,*] buffers

typedef __attribute__((ext_vector_type(2))) float v2f;
typedef __attribute__((ext_vector_type(8))) float v8f;

// ---------------- utility kernels ----------------

__global__ void arma_zero(float* __restrict__ p, int n) {
  int i = blockIdx.x * blockDim.x + threadIdx.x;
  if (i < n) p[i] = 0.0f;
}

__global__ void arma_deg(const int* __restrict__ ei, float* __restrict__ deg) {
  int e = blockIdx.x * blockDim.x + threadIdx.x;
  if (e < N_EDGES) atomicAdd(&deg[ei[N_EDGES + e]], 1.0f);  // col = ei[1][e]
}

__global__ void arma_dis(float* __restrict__ deg) {
  int n = blockIdx.x * blockDim.x + threadIdx.x;
  if (n < N_NODES) {
    float d = deg[n];
    deg[n] = (d > 0.0f) ? rsqrtf(d) : 0.0f;   // deg>=1 when >0, max(deg,1)=deg
  }
}

__global__ void arma_norm(const int* __restrict__ ei, const float* __restrict__ dis,
                          float* __restrict__ norm) {
  int e = blockIdx.x * blockDim.x + threadIdx.x;
  if (e < N_EDGES) norm[e] = dis[ei[e]] * dis[ei[N_EDGES + e]];
}

// ---------------- WMMA GEMMs (fp32, 16x16x4), LDS-staged ----------------
// A fragment (16x4): lane l -> row M = l&15; lanes 0-15 carry K=kk+0/kk+1,
// lanes 16-31 carry K=kk+2/kk+3 (koff = (l>>4)*2).
// B fragment (4x16): same K split across lane halves, col N = l&15.
// C/D (16x16): VGPR r, lanes 0-15 -> M=r, lanes 16-31 -> M=8+r, N = l&15.

// Fused root+init GEMM: one wave per 16-node tile computes all 2K column tiles
// from a single LDS-staged A tile. out[k,n,:] = X[n,:] @ W[k] (+bias for root),
// stored zero-padded to 16 columns.
__global__ void arma_gemm_in(const float* __restrict__ X, int Fin,
                             const float* __restrict__ Wroot,
                             const float* __restrict__ Winit,
                             const float* __restrict__ bias, int Fout,
                             float* __restrict__ root_out,
                             float* __restrict__ t_out) {
  __shared__ __align__(16) float ldsX[16 * F_INF];   // 8 KB max (A tile, row-major)
  __shared__ __align__(16) float ldsW[F_INF * 16];   // 8 KB max (B tile, padded)

  const int l    = threadIdx.x;                 // 0..31 (wave32)
  const int m0   = blockIdx.x * 16;             // N_NODES % 16 == 0
  const int col  = l & 15;
  const int koff = (l >> 4) * 2;

  // Stage A tile: rows m0..m0+15 of X are contiguous -> coalesced float4 copy.
  {
    const float4* s4 = (const float4*)(X + (size_t)m0 * Fin);
    float4* d4 = (float4*)ldsX;
    const int nv = (16 * Fin) >> 2;
    for (int i = l; i < nv; i += 32) d4[i] = s4[i];
  }

  for (int c = 0; c < 2 * K_ST; ++c) {
    const bool isRoot = (c < K_ST);
    const int k = isRoot ? c : c - K_ST;
    const float* W = (isRoot ? Wroot : Winit) + (size_t)k * Fin * Fout;

    __syncthreads();   // A staged / previous B consumers done
    // Stage zero-padded B tile [Fin][16]; clamped address keeps load in-bounds,
    // select handles padding -> no divergent exec in the hot loop.
    for (int i = l; i < Fin * 16; i += 32) {
      int row = i >> 4, cc = i & 15;
      float v = W[(cc < Fout) ? (row * Fout + cc) : 0];
      ldsW[i] = (cc < Fout) ? v : 0.0f;
    }
    __syncthreads();

    v8f acc = {};
    for (int kk = 0; kk < Fin; kk += 4) {
      const int r0 = kk + koff;
      v2f a, b;
      a.x = ldsX[col * Fin + r0];
      a.y = ldsX[col * Fin + r0 + 1];
      b.x = ldsW[(r0 << 4) | col];
      b.y = ldsW[((r0 + 1) << 4) | col];
      acc = __builtin_amdgcn_wmma_f32_16x16x4_f32(false, a, false, b,
                                                  (short)0, acc, false, false);
    }

    float bv = (isRoot && col < Fout) ? bias[k * Fout + col] : 0.0f;
    float* dst = isRoot ? root_out : t_out;
    const int hi = (l >> 4) * 8;
#pragma unroll
    for (int r = 0; r < 8; ++r) {
      int mm = m0 + r + hi;
      dst[((size_t)k * N_NODES + mm) * PADW + col] = acc[r] + bv;
    }
  }
}

// Stack GEMM: y[k,n,:] = act(prop[k,n,:] + root[k,n,:]) @ Wstack[k] (Wd x Wd, padded)
__global__ void arma_gemm_stack(const float* __restrict__ prop,
                                const float* __restrict__ root,
                                const float* __restrict__ Wstack, int Wd,
                                int do_relu, float* __restrict__ t_out) {
  __shared__ __align__(16) float ldsZ[256];   // act(prop+root) tile [16][16]
  __shared__ __align__(16) float ldsW[256];   // padded weight [16][16]

  const int l    = threadIdx.x;
  const int m0   = blockIdx.x * 16;
  const int k    = blockIdx.y;
  const float* W = Wstack + (size_t)k * Wd * Wd;
  const int col  = l & 15;
  const int koff = (l >> 4) * 2;

  // Stage padded weight tile.
  for (int i = l; i < 256; i += 32) {
    int row = i >> 4, cc = i & 15;
    bool ok = (row < Wd) && (cc < Wd);
    float v = W[ok ? (row * Wd + cc) : 0];
    ldsW[i] = ok ? v : 0.0f;
  }
  // Stage activation tile (contiguous -> coalesced).
  const size_t tbase = ((size_t)k * N_NODES + m0) * PADW;
  for (int i = l; i < 256; i += 32) {
    float z = prop[tbase + i] + root[tbase + i];
    if (do_relu) z = fmaxf(z, 0.0f);
    ldsZ[i] = z;
  }
  __syncthreads();

  v8f acc = {};
#pragma unroll
  for (int kk = 0; kk < PADW; kk += 4) {
    const int r0 = kk + koff;
    v2f a, b;
    a.x = ldsZ[(col << 4) | r0];
    a.y = ldsZ[(col << 4) | (r0 + 1)];
    b.x = ldsW[(r0 << 4) | col];
    b.y = ldsW[((r0 + 1) << 4) | col];
    acc = __builtin_amdgcn_wmma_f32_16x16x4_f32(false, a, false, b,
                                                (short)0, acc, false, false);
  }
  const int hi = (l >> 4) * 8;
#pragma unroll
  for (int r = 0; r < 8; ++r) {
    int mm = m0 + r + hi;
    t_out[((size_t)k * N_NODES + mm) * PADW + col] = acc[r];
  }
}

// ---------------- edge propagation: dst[k,col,:] += src[k,row,:] * norm[e] ----------------
// Destination [K,N,16] = 19.2 MB -> L2-resident on MI455X; scatter resolves at L2 atomics.
__global__ void arma_propagate(const int* __restrict__ ei,
                               const float* __restrict__ norm,
                               const float* __restrict__ src,
                               float* __restrict__ dst) {
  int i = blockIdx.x * blockDim.x + threadIdx.x;      // i = k*E + e
  if (i >= K_ST * N_EDGES) return;
  int e = i % N_EDGES;
  int k = i / N_EDGES;
  int r = ei[e];
  int c = ei[N_EDGES + e];
  float w = norm[e];
  const float4* s = (const float4*)(src + ((size_t)k * N_NODES + r) * PADW);
  float* d = dst + ((size_t)k * N_NODES + c) * PADW;
#pragma unroll
  for (int j = 0; j < 4; ++j) {
    float4 v = s[j];
    atomicAdd(d + 4 * j + 0, v.x * w);
    atomicAdd(d + 4 * j + 1, v.y * w);
    atomicAdd(d + 4 * j + 2, v.z * w);
    atomicAdd(d + 4 * j + 3, v.w * w);
  }
}

// h[n,f] = mean_k relu(prop[k,n,f] + root[k,n,f])
__global__ void arma_hmean(const float* __restrict__ prop,
                           const float* __restrict__ root,
                           float* __restrict__ h) {
  int i = blockIdx.x * blockDim.x + threadIdx.x;
  if (i < N_NODES * PADW) {
    float s = 0.0f;
#pragma unroll
    for (int k = 0; k < K_ST; ++k) {
      size_t idx = (size_t)k * N_NODES * PADW + i;
      s += fmaxf(prop[idx] + root[idx], 0.0f);
    }
    h[i] = s * (1.0f / (float)K_ST);
  }
}

// out[n,o] = mean_k (prop[k,n,o] + root[k,n,o]), o < 10, no relu
__global__ void arma_final(const float* __restrict__ prop,
                           const float* __restrict__ root,
                           float* __restrict__ out) {
  int i = blockIdx.x * blockDim.x + threadIdx.x;
  if (i < N_NODES * OUT_F) {
    int n = i / OUT_F, o = i % OUT_F;
    size_t idx = (size_t)n * PADW + o;
    float s = 0.0f;
#pragma unroll
    for (int k = 0; k < K_ST; ++k) {
      size_t p = (size_t)k * N_NODES * PADW + idx;
      s += prop[p] + root[p];
    }
    out[i] = s * (1.0f / (float)K_ST);
  }
}

// ---------------- host launch ----------------

extern "C" void kernel_launch(void* const* d_in, const int* in_sizes, int n_in,
                              void* d_out, int out_size, void* d_ws, size_t ws_size,
                              hipStream_t stream) {
  (void)in_sizes; (void)n_in; (void)out_size; (void)ws_size;
  const float* x        = (const float*)d_in[0];
  const int*   ei       = (const int*)  d_in[1];
  const float* w1_init  = (const float*)d_in[2];
  const float* w1_stack = (const float*)d_in[3];
  const float* w1_root  = (const float*)d_in[4];
  const float* w1_bias  = (const float*)d_in[5];
  const float* w2_init  = (const float*)d_in[6];
  const float* w2_stack = (const float*)d_in[7];
  const float* w2_root  = (const float*)d_in[8];
  const float* w2_bias  = (const float*)d_in[9];
  float* out = (float*)d_out;

  // workspace partition (floats), ~71 MB total
  const size_t KNF = (size_t)K_ST * N_NODES * PADW;   // 4.8M
  float* ws   = (float*)d_ws;
  float* deg  = ws;                    // N (becomes dis in place)
  float* norm = deg  + N_NODES;        // E
  float* root = norm + N_EDGES;        // KNF (reused by conv1 & conv2)
  float* tbuf = root + KNF;            // KNF
  float* prop = tbuf + KNF;            // KNF
  float* h    = prop + KNF;            // N*16

  const int ZB = 256;
  const int nTiles = N_NODES / 16;     // 6250, exact

  // 1. gcn_norm
  arma_zero<<<(N_NODES + ZB - 1) / ZB, ZB, 0, stream>>>(deg, N_NODES);
  arma_deg<<<(N_EDGES + ZB - 1) / ZB, ZB, 0, stream>>>(ei, deg);
  arma_dis<<<(N_NODES + ZB - 1) / ZB, ZB, 0, stream>>>(deg);
  arma_norm<<<(N_EDGES + ZB - 1) / ZB, ZB, 0, stream>>>(ei, deg, norm);

  const int propThreads = K_ST * N_EDGES;
  const int KNFi = (int)KNF;

  // 2. conv1: fused root+init GEMM (WMMA from LDS)
  arma_gemm_in<<<nTiles, 32, 0, stream>>>(
      x, F_INF, w1_root, w1_init, w1_bias, H_F, root, tbuf);
  // propagate init
  arma_zero<<<(KNFi + ZB - 1) / ZB, ZB, 0, stream>>>(prop, KNFi);
  arma_propagate<<<(propThreads + ZB - 1) / ZB, ZB, 0, stream>>>(ei, norm, tbuf, prop);
  // stack GEMM with fused relu(prop+root)
  arma_gemm_stack<<<dim3(nTiles, K_ST), 32, 0, stream>>>(
      prop, root, w1_stack, H_F, 1, tbuf);
  // propagate stack output
  arma_zero<<<(KNFi + ZB - 1) / ZB, ZB, 0, stream>>>(prop, KNFi);
  arma_propagate<<<(propThreads + ZB - 1) / ZB, ZB, 0, stream>>>(ei, norm, tbuf, prop);
  // h = mean_k relu(prop+root)
  arma_hmean<<<(N_NODES * PADW + ZB - 1) / ZB, ZB, 0, stream>>>(prop, root, h);

  // 3. conv2 (reuse root/tbuf/prop)
  arma_gemm_in<<<nTiles, 32, 0, stream>>>(
      h, H_F, w2_root, w2_init, w2_bias, OUT_F, root, tbuf);
  arma_zero<<<(KNFi + ZB - 1) / ZB, ZB, 0, stream>>>(prop, KNFi);
  arma_propagate<<<(propThreads + ZB - 1) / ZB, ZB, 0, stream>>>(ei, norm, tbuf, prop);
  arma_gemm_stack<<<dim3(nTiles, K_ST), 32, 0, stream>>>(
      prop, root, w2_stack, OUT_F, 0, tbuf);
  arma_zero<<<(KNFi + ZB - 1) / ZB, ZB, 0, stream>>>(prop, KNFi);
  arma_propagate<<<(propThreads + ZB - 1) / ZB, ZB, 0, stream>>>(ei, norm, tbuf, prop);
  // 4. output mean (no relu)
  arma_final<<<(N_NODES * OUT_F + ZB - 1) / ZB, ZB, 0, stream>>>(prop, root, out);
}